// MambaSelectiveScan_89489938580221
// MI455X (gfx1250) — compile-verified
//
#include <hip/hip_runtime.h>
#include <math.h>

#define DD 2048      // model dim
#define NS 16        // state dim N
#define RR 128       // rank R = D/N
#define EE 144       // N + R (GEMM1 output width)
#define LL 2048      // sequence length
#define NCH 16       // scan chunks
#define CLEN 128     // timesteps per chunk  (NCH*CLEN == LL)

typedef float v2f __attribute__((ext_vector_type(2)));
typedef float v8f __attribute__((ext_vector_type(8)));

// ---------------------------------------------------------------------------
// GEMM 1: dB[l][e] = sum_d X[l][d] * W_in[e][d]      (L x D) * (E x D)^T
// One 16x16 output tile per wave, V_WMMA_F32_16X16X4_F32, K-loop over D.
// A lane layout (ISA 7.12.2, 32-bit A 16x4): lane = (k>=2)*16 + m, vgpr = k&1
// B symmetric: lane = (k>=2)*16 + n, so each lane loads contiguous float2s.
// ---------------------------------------------------------------------------
__global__ __launch_bounds__(256) void gemm_in_kernel(const float* __restrict__ X,
                                                      const float* __restrict__ W,
                                                      float* __restrict__ dB) {
  const int lane  = threadIdx.x & 31;
  const int wave  = threadIdx.x >> 5;
  const int tile  = blockIdx.x * 8 + wave;       // 0..1151
  const int ntile = tile % (EE / 16);            // 0..8
  const int mtile = tile / (EE / 16);            // 0..127
  const int m16   = lane & 15;
  const int khalf = lane >> 4;                   // 0 or 1 -> K pair {0,1} or {2,3}

  const float* arow = X + (size_t)(mtile * 16 + m16) * DD + khalf * 2;
  const float* brow = W + (size_t)(ntile * 16 + m16) * DD + khalf * 2;

  v8f acc = {};
  for (int k = 0; k < DD; k += 4) {
    v2f a = *(const v2f*)(arow + k);
    v2f b = *(const v2f*)(brow + k);
    acc = __builtin_amdgcn_wmma_f32_16x16x4_f32(false, a, false, b,
                                                (short)0, acc, false, false);
  }
  // C/D layout: VGPR v holds M = v + 8*khalf, N = lane&15
  const int n_out  = ntile * 16 + m16;
  const int m_base = mtile * 16 + khalf * 8;
#pragma unroll
  for (int v = 0; v < 8; ++v)
    dB[(size_t)(m_base + v) * EE + n_out] = acc[v];
}

// ---------------------------------------------------------------------------
// Per-row RMS norm of dB: cols [0,16) with B_ln_w over 16, cols [16,144)
// with dt_ln_w over 128. In place. 128 threads per row, LDS reductions.
// ---------------------------------------------------------------------------
__global__ __launch_bounds__(128) void rmsnorm_kernel(float* __restrict__ dB,
                                                      const float* __restrict__ B_ln_w,
                                                      const float* __restrict__ dt_ln_w) {
  __shared__ float redD[128];
  __shared__ float redB[16];
  const int t = threadIdx.x;
  float* row = dB + (size_t)blockIdx.x * EE;

  float dv = row[NS + t];
  redD[t] = dv * dv;
  float bv = 0.f;
  if (t < NS) { bv = row[t]; redB[t] = bv * bv; }
  __syncthreads();

  for (int s = 64; s > 0; s >>= 1) {
    if (t < s) redD[t] += redD[t + s];
    __syncthreads();
  }
  if (t < 8) redB[t] += redB[t + 8];
  __syncthreads();
  if (t < 4) redB[t] += redB[t + 4];
  __syncthreads();
  if (t < 2) redB[t] += redB[t + 2];
  __syncthreads();
  if (t < 1) redB[0] += redB[1];
  __syncthreads();

  const float dscale = rsqrtf(redD[0] * (1.0f / RR) + 1e-5f);
  row[NS + t] = dv * dscale * (dt_ln_w[t] + 1.0f);
  if (t < NS) {
    const float bscale = rsqrtf(redB[0] * (1.0f / NS) + 1e-5f);
    row[t] = bv * bscale * (B_ln_w[t] + 1.0f);
  }
}

// ---------------------------------------------------------------------------
// GEMM 2 + epilogue: delta[l][d] = softplus( sum_r dBn[l][16+r]*dt_w[d][r]
//                                            + dt_b[d] )
// (L x R) * (D x R)^T, K = 128 -> 32 WMMAs per tile.
// ---------------------------------------------------------------------------
__global__ __launch_bounds__(256) void gemm_dt_kernel(const float* __restrict__ dBn,
                                                      const float* __restrict__ dt_w,
                                                      const float* __restrict__ dt_b,
                                                      float* __restrict__ delta) {
  const int lane  = threadIdx.x & 31;
  const int wave  = threadIdx.x >> 5;
  const int tile  = blockIdx.x * 8 + wave;       // 0..16383
  const int dtile = tile % (DD / 16);            // 0..127 over D
  const int ltile = tile / (DD / 16);            // 0..127 over L
  const int m16   = lane & 15;
  const int khalf = lane >> 4;

  const float* arow = dBn  + (size_t)(ltile * 16 + m16) * EE + NS + khalf * 2;
  const float* brow = dt_w + (size_t)(dtile * 16 + m16) * RR + khalf * 2;

  v8f acc = {};
  for (int k = 0; k < RR; k += 4) {
    v2f a = *(const v2f*)(arow + k);
    v2f b = *(const v2f*)(brow + k);
    acc = __builtin_amdgcn_wmma_f32_16x16x4_f32(false, a, false, b,
                                                (short)0, acc, false, false);
  }
  const int d      = dtile * 16 + m16;
  const int l_base = ltile * 16 + khalf * 8;
  const float bias = dt_b[d];
#pragma unroll
  for (int v = 0; v < 8; ++v) {
    float x  = acc[v] + bias;
    float sp = (x > 20.0f) ? x : log1pf(__expf(x));   // softplus
    delta[(size_t)(l_base + v) * DD + d] = sp;
  }
}

// ---------------------------------------------------------------------------
// Scan phase A: per (chunk c, d, n), run 128 sequential steps
//   dA = exp(delta*A[d][n]); s = dA*s + delta*x*B[l][n]; aprod *= dA
// gid = c*32768 + n*2048 + d  -> consecutive threads read consecutive d
// (coalesced delta/X), B[l][n] is a broadcast, A = -exp(A_log).
// ---------------------------------------------------------------------------
__global__ __launch_bounds__(256) void scan_chunks_kernel(const float* __restrict__ delta,
                                                          const float* __restrict__ X,
                                                          const float* __restrict__ dBn,
                                                          const float* __restrict__ A_log,
                                                          float* __restrict__ chunkA,
                                                          float* __restrict__ chunkS) {
  const int gid = blockIdx.x * 256 + threadIdx.x;  // 0..524287
  const int c   = gid >> 15;                       // / (D*N)
  const int rem = gid & 32767;
  const int n   = rem >> 11;                       // / D
  const int d   = rem & 2047;

  const float Adn = -__expf(A_log[d * NS + n]);
  float aprod = 1.0f, s = 0.0f;
  const int l0 = c * CLEN;
  for (int i = 0; i < CLEN; ++i) {
    const int l = l0 + i;
    __builtin_prefetch(&delta[(size_t)(l + 8) * DD + d], 0, 0);  // global_prefetch_b8
    const float dlt = delta[(size_t)l * DD + d];
    const float xv  = X[(size_t)l * DD + d];
    const float Bn  = dBn[(size_t)l * EE + n];
    const float dA  = __expf(dlt * Adn);
    s = fmaf(dA, s, dlt * xv * Bn);
    aprod *= dA;
  }
  chunkA[gid] = aprod;
  chunkS[gid] = s;
}

// ---------------------------------------------------------------------------
// Scan phase B: fold the 16 chunk (a,b) pairs in order; out[d*16+n].
// ---------------------------------------------------------------------------
__global__ __launch_bounds__(256) void scan_combine_kernel(const float* __restrict__ chunkA,
                                                           const float* __restrict__ chunkS,
                                                           float* __restrict__ out) {
  const int idx = blockIdx.x * 256 + threadIdx.x;  // n*2048 + d
  const int n = idx >> 11;
  const int d = idx & 2047;
  float s = 0.0f;
#pragma unroll
  for (int c = 0; c < NCH; ++c)
    s = chunkA[c * (DD * NS) + idx] * s + chunkS[c * (DD * NS) + idx];
  out[d * NS + n] = s;
}

// ---------------------------------------------------------------------------
// Launch. Inputs (setup_inputs order, all f32):
//  0 hidden_states (1,2048,2048)  1 W_in (144,2048)  2 dt_w (2048,128)
//  3 dt_b (2048)  4 A_log (2048,16)  5 dt_ln_w (128)  6 B_ln_w (16)
// Output: (1,2048,16) f32. Workspace usage ~22.2 MB.
// ---------------------------------------------------------------------------
extern "C" void kernel_launch(void* const* d_in, const int* in_sizes, int n_in,
                              void* d_out, int out_size, void* d_ws, size_t ws_size,
                              hipStream_t stream) {
  const float* X       = (const float*)d_in[0];
  const float* W_in    = (const float*)d_in[1];
  const float* dt_w    = (const float*)d_in[2];
  const float* dt_b    = (const float*)d_in[3];
  const float* A_log   = (const float*)d_in[4];
  const float* dt_ln_w = (const float*)d_in[5];
  const float* B_ln_w  = (const float*)d_in[6];
  float* out = (float*)d_out;

  float* ws     = (float*)d_ws;
  float* dB     = ws;                               // 2048*144      = 294912 f
  float* delta  = dB + (size_t)LL * EE;             // 2048*2048     = 4194304 f
  float* chunkA = delta + (size_t)LL * DD;          // 16*2048*16    = 524288 f
  float* chunkS = chunkA + (size_t)NCH * DD * NS;   // 524288 f

  // 1) dB = X @ W_in^T : 1152 tiles, 8 waves/block
  gemm_in_kernel<<<dim3((LL / 16) * (EE / 16) / 8), 256, 0, stream>>>(X, W_in, dB);
  // 2) per-row RMS norm (in place)
  rmsnorm_kernel<<<dim3(LL), 128, 0, stream>>>(dB, B_ln_w, dt_ln_w);
  // 3) delta = softplus(dBn[:,16:] @ dt_w^T + dt_b) : 16384 tiles
  gemm_dt_kernel<<<dim3((LL / 16) * (DD / 16) / 8), 256, 0, stream>>>(dB, dt_w, dt_b, delta);
  // 4) chunked scan: 16 chunks x 32768 (d,n) pairs
  scan_chunks_kernel<<<dim3(NCH * DD * NS / 256), 256, 0, stream>>>(delta, X, dB, A_log,
                                                                    chunkA, chunkS);
  // 5) fold chunks, write final state
  scan_combine_kernel<<<dim3(DD * NS / 256), 256, 0, stream>>>(chunkA, chunkS, out);
}